// NEDT_283467842025
// MI455X (gfx1250) — compile-verified
//
#include <hip/hip_runtime.h>
#include <hip/hip_bf16.h>
#include <math.h>

typedef __attribute__((ext_vector_type(2))) float v2f;
typedef __attribute__((ext_vector_type(8))) float v8f;

constexpr int BN = 16, CH = 3, HN = 544, WN = 960;
constexpr int NPIX = BN * HN * WN;         // 8,355,840
constexpr float BIGF = 1000000.0f;

struct GW { float w0[9]; float w1[13]; };

// ---------------- 1) RGB -> gray ----------------
__global__ void __launch_bounds__(256) k_gray(const float* __restrict__ img,
                                              float* __restrict__ gray) {
  int i = blockIdx.x * 256 + threadIdx.x;          // [b,y,x] flat, exact grid
  int b = i / (HN * WN);
  int p = i - b * HN * WN;
  const float* base = img + (size_t)b * CH * HN * WN;
  float r  = base[p];
  float g  = base[HN * WN + p];
  float bl = base[2 * HN * WN + p];
  gray[i] = 0.299f * r + 0.587f * g + 0.114f * bl;
}

// ---------------- 2) vertical blur (both sigmas), replicate-y ----------------
__global__ void __launch_bounds__(256) k_vblur(const float* __restrict__ gray,
                                               float* __restrict__ t0,
                                               float* __restrict__ t1, GW gw) {
  int i = blockIdx.x * 256 + threadIdx.x;
  int b = i / (HN * WN);
  int p = i - b * HN * WN;
  int y = p / WN;
  int x = p - y * WN;
  const float* gb = gray + b * HN * WN;
  float a0 = 0.f;
#pragma unroll
  for (int t = 0; t < 9; ++t) {
    int yy = y + t - 4; yy = yy < 0 ? 0 : (yy > HN - 1 ? HN - 1 : yy);
    a0 += gw.w0[t] * gb[yy * WN + x];
  }
  float a1 = 0.f;
#pragma unroll
  for (int t = 0; t < 13; ++t) {
    int yy = y + t - 6; yy = yy < 0 ? 0 : (yy > HN - 1 ? HN - 1 : yy);
    a1 += gw.w1[t] * gb[yy * WN + x];
  }
  t0[i] = a0;
  t1[i] = a1;
}

// ---------------- 3) horizontal blur via FP32 WMMA + DoG threshold ----------
// One wave computes one 16x16 output tile. out = A(16x28) * B(28x16), where
// A = vblurred rows (x-replicate clamped), B = banded Gaussian weights.
// 28 = 7 chunks of K=4 -> 7 WMMAs per blur, two blurs -> 14 WMMAs per tile.
__global__ void __launch_bounds__(256)
k_hblur_wmma(const float* __restrict__ t0, const float* __restrict__ t1,
             float* __restrict__ mask,
             float inv2s0, float isum0, float inv2s1, float isum1) {
  const int lane = threadIdx.x & 31;
  const int wave = threadIdx.x >> 5;
  int tile = blockIdx.x * 8 + wave;                 // 4080 blocks * 8 = 32640
  int b   = tile / (34 * 60);
  int rem = tile - b * 34 * 60;
  int ty  = rem / 60;
  int tx  = rem - ty * 60;
  int y0 = ty * 16, x0 = tx * 16, s = x0 - 6;       // window start (r1 = 6)

  int hi = lane >> 4;                               // K half select
  int n  = lane & 15;                               // A row / B col
  int rowBase = (b * HN + (y0 + n)) * WN;

  // ---- Precompute tile-invariant band-matrix B values (per lane) ----
  // B1[j,n] = w1[j-n] for 0<=j-n<=12 ; B0[j,n] = w0[j-n-2] for 2<=j-n<=10,
  // both centered at j-n == 6. Done before the WMMA stream so exp's
  // special-case divergence never interleaves with the matrix ops.
  v2f bw0[7], bw1[7];
#pragma unroll
  for (int c = 0; c < 7; ++c) {
    int j0 = 4 * c + 2 * hi;
    int dj0 = j0 - n, dj1 = j0 + 1 - n;
    float f0 = (float)(dj0 - 6), f1 = (float)(dj1 - 6);
    float e0 = f0 * f0, e1 = f1 * f1;
    bw1[c].x = (dj0 >= 0 && dj0 <= 12) ? expf(-e0 * inv2s1) * isum1 : 0.f;
    bw1[c].y = (dj1 >= 0 && dj1 <= 12) ? expf(-e1 * inv2s1) * isum1 : 0.f;
    bw0[c].x = (dj0 >= 2 && dj0 <= 10) ? expf(-e0 * inv2s0) * isum0 : 0.f;
    bw0[c].y = (dj1 >= 2 && dj1 <= 10) ? expf(-e1 * inv2s0) * isum0 : 0.f;
  }

  v8f d0 = {0.f, 0.f, 0.f, 0.f, 0.f, 0.f, 0.f, 0.f};
  v8f d1 = {0.f, 0.f, 0.f, 0.f, 0.f, 0.f, 0.f, 0.f};

#pragma unroll
  for (int c = 0; c < 7; ++c) {
    int j0 = 4 * c + 2 * hi;                        // K index of .x component
    int c0 = s + j0, c1 = s + j0 + 1;               // replicate-x clamp
    c0 = c0 < 0 ? 0 : (c0 > WN - 1 ? WN - 1 : c0);
    c1 = c1 < 0 ? 0 : (c1 > WN - 1 ? WN - 1 : c1);

    v2f a0, a1;
    a0.x = t0[rowBase + c0]; a0.y = t0[rowBase + c1];
    a1.x = t1[rowBase + c0]; a1.y = t1[rowBase + c1];

    d0 = __builtin_amdgcn_wmma_f32_16x16x4_f32(false, a0, false, bw0[c],
                                               (short)0, d0, false, false);
    d1 = __builtin_amdgcn_wmma_f32_16x16x4_f32(false, a1, false, bw1[c],
                                               (short)0, d1, false, false);
  }

  // C/D layout: VGPR v -> M = v (lanes 0-15) or v+8 (lanes 16-31), N = lane&15
#pragma unroll
  for (int v = 0; v < 8; ++v) {
    int orow = y0 + v + 8 * hi;
    float diff = d1[v] - d0[v];                     // dog>0.5 <=> g1-g0>0.005
    mask[(b * HN + orow) * WN + x0 + n] = diff > 0.005f ? 1.0f : 0.0f;
  }
}

// ---------------- 4) per-row 1D EDT (squared) ----------------
__global__ void __launch_bounds__(256) k_rowedt(const float* __restrict__ mask,
                                                float* __restrict__ g) {
  int tid = blockIdx.x * 256 + threadIdx.x;         // 34*256 = 8704 rows exact
  int base = tid * WN;
  float last = -BIGF;                               // left = cummax(fg? idx : -BIG)
  for (int x = 0; x < WN; ++x) {
    if ((x & 63) == 0 && x + 64 < WN)
      __builtin_prefetch(mask + base + x + 64, 0, 1);   // global_prefetch_b8
    float fx = (float)x;
    if (mask[base + x] > 0.5f) last = fx;
    g[base + x] = fx - last;                        // idx - left
  }
  float next = BIGF;                                // right = cummin(fg? idx : BIG)
  for (int x = WN - 1; x >= 0; --x) {
    float fx = (float)x;
    if (mask[base + x] > 0.5f) next = fx;
    float d = fminf(g[base + x], next - fx);
    g[base + x] = d * d;
  }
}

// ---------------- 5) column lower-envelope + final transform ----------------
// Tile fill uses the gfx1250 async global->LDS engine (ASYNCcnt-tracked),
// then the O(H^2) min-plus envelope runs out of LDS on VALU.
__global__ void __launch_bounds__(256) k_coledt(const float* __restrict__ g,
                                                float* __restrict__ out) {
  __shared__ float sg[HN * 16];
  int x0 = blockIdx.x * 16;                         // 60 x-tiles
  int b  = blockIdx.y;                              // 16 batches
  int tid = threadIdx.x;

  unsigned long long gbase = (unsigned long long)(const void*)g;
#pragma unroll 2
  for (int i = tid; i < HN * 16; i += 256) {
    int ys = i >> 4, c = i & 15;
    unsigned voff = (unsigned)(((b * HN + ys) * WN + x0 + c) * 4);
    // Flat->LDS mapping truncates to addr[31:0]: low 32 bits of the generic
    // pointer to shared memory are the LDS byte address for VDST.
    unsigned lds = (unsigned)(unsigned long long)(const void*)&sg[i];
    asm volatile("global_load_async_to_lds_b32 %0, %1, %2 offset:0"
                 :: "v"(lds), "v"(voff), "s"(gbase) : "memory");
  }
  asm volatile("s_wait_asynccnt 0x0" ::: "memory");
  __syncthreads();

  int tx = tid & 15, tyl = tid >> 4;
  for (int yt = 0; yt < 34; ++yt) {
    int y = tyl * 34 + yt;
    float m = 3.4e38f;
    for (int ys = 0; ys < HN; ++ys) {
      float fd = (float)(y - ys);
      float v = sg[ys * 16 + tx] + fd * fd;         // fd*fd integer-exact
      m = fminf(m, v);
    }
    out[(b * HN + y) * WN + x0 + tx] = 1.0f - expf(-sqrtf(m) * 0.0375f);
  }
}

extern "C" void kernel_launch(void* const* d_in, const int* in_sizes, int n_in,
                              void* d_out, int out_size, void* d_ws, size_t ws_size,
                              hipStream_t stream) {
  (void)in_sizes; (void)n_in; (void)out_size; (void)ws_size;
  const float* img = (const float*)d_in[0];
  float* out  = (float*)d_out;
  float* gray = (float*)d_ws;          // reused as g after mask is built
  float* t0   = gray + NPIX;
  float* t1   = t0 + NPIX;

  // Gaussian weights (host): sigma0 = 544/540, sigma1 = 1.6*sigma0
  // kern0 = 2*int(4*s0)+1 = 9, kern1 = 2*int(4*s1)+1 = 13
  GW gw;
  double s0 = 544.0 / 540.0, s1 = s0 * 1.6;
  double e0[9], e1[13], sum0 = 0.0, sum1 = 0.0;
  for (int i = 0; i < 9; ++i)  { double xv = i - 4; e0[i] = exp(-(xv*xv)/(2.0*s0*s0)); sum0 += e0[i]; }
  for (int i = 0; i < 13; ++i) { double xv = i - 6; e1[i] = exp(-(xv*xv)/(2.0*s1*s1)); sum1 += e1[i]; }
  for (int i = 0; i < 9; ++i)  gw.w0[i] = (float)(e0[i] / sum0);
  for (int i = 0; i < 13; ++i) gw.w1[i] = (float)(e1[i] / sum1);
  float inv2s0 = (float)(1.0 / (2.0 * s0 * s0));
  float inv2s1 = (float)(1.0 / (2.0 * s1 * s1));
  float isum0  = (float)(1.0 / sum0);
  float isum1  = (float)(1.0 / sum1);

  k_gray <<<NPIX / 256, 256, 0, stream>>>(img, gray);
  k_vblur<<<NPIX / 256, 256, 0, stream>>>(gray, t0, t1, gw);
  k_hblur_wmma<<<4080, 256, 0, stream>>>(t0, t1, out /*mask scratch*/,
                                         inv2s0, isum0, inv2s1, isum1);
  k_rowedt<<<34, 256, 0, stream>>>(out /*mask*/, gray /*g*/);
  k_coledt<<<dim3(60, 16), 256, 0, stream>>>(gray /*g*/, out);
}